// GCN_9869834846342
// MI455X (gfx1250) — compile-verified
//
#include <hip/hip_runtime.h>
#include <math.h>

typedef __attribute__((ext_vector_type(2))) float v2f;
typedef __attribute__((ext_vector_type(8))) float v8f;

#define F 128          // feature width (F_IN == D_HID == 128)
#define NCLS 40        // output classes

// ---------------------------------------------------------------------------
// degree / normalization
// ---------------------------------------------------------------------------
__global__ void init_deg_k(float* __restrict__ deg, int n) {
    int i = blockIdx.x * blockDim.x + threadIdx.x;
    if (i < n) deg[i] = 1.0f;                      // self-loop contributes 1
}

__global__ void deg_edges_k(const int* __restrict__ ei, float* __restrict__ deg, int E) {
    int e = blockIdx.x * blockDim.x + threadIdx.x;
    if (e < E) atomicAdd(&deg[ei[E + e]], 1.0f);   // dst = ei[1][e]
}

__global__ void dinv_k(const float* __restrict__ deg, float* __restrict__ dinv, int n) {
    int i = blockIdx.x * blockDim.x + threadIdx.x;
    if (i < n) {
        float d = deg[i];
        dinv[i] = (d > 0.0f) ? rsqrtf(d) : 0.0f;
    }
}

__global__ void norm_k(const int* __restrict__ ei, const float* __restrict__ dinv,
                       float* __restrict__ norm, int E, int total) {
    int e = blockIdx.x * blockDim.x + threadIdx.x;
    if (e >= total) return;
    if (e < E) {
        norm[e] = dinv[ei[e]] * dinv[ei[E + e]];
    } else {
        float d = dinv[e - E];
        norm[e] = d * d;                            // self-loop
    }
}

// ---------------------------------------------------------------------------
// WMMA GEMM: Out[N,128] = H[N,128] @ W[128,128]   (full f32, 16x16x4 WMMA)
// block = 128 threads (4 waves), each block covers 64 rows; wave w -> 16 rows.
// Weights staged in LDS pair-interleaved: Wlds[p*128+n] = {W[2p][n], W[2p+1][n]}
// so a B fragment is one aligned ds_load_b64 (stride-8B across lanes: no bank
// conflicts, no register shuffling).
// A rows are clamped instead of zero-filled: A row M only feeds C row M, and
// OOB C rows are never stored, so no exec masking is needed in the hot loop.
// ---------------------------------------------------------------------------
__global__ void __launch_bounds__(128) gemm128_wmma_k(const float* __restrict__ H,
                                                      const float* __restrict__ W,
                                                      float* __restrict__ Out,
                                                      int nrows) {
    __shared__ float2 Wlds[(F / 2) * F];            // 64 KB, pair-interleaved
    for (int i = threadIdx.x; i < (F / 2) * F; i += 128) {
        const int p = i >> 7;                       // k-pair index 0..63
        const int n = i & (F - 1);                  // column 0..127
        Wlds[i] = make_float2(W[(2 * p) * F + n], W[(2 * p + 1) * F + n]);
    }
    __syncthreads();

    const int tid  = threadIdx.x;
    const int lane = tid & 31;
    const int half = lane >> 4;                     // 0 or 1
    const int l    = lane & 15;                     // 0..15
    const int wave = tid >> 5;
    const int m0   = blockIdx.x * 64 + wave * 16;   // first row of this wave's strip

    v8f acc[8];
#pragma unroll
    for (int nt = 0; nt < 8; ++nt) { v8f z = {}; acc[nt] = z; }

    const int row  = m0 + l;                        // A-matrix row for this lane
    const int arow = (row < nrows) ? row : (nrows - 1);   // clamp (see note above)
    const float* hrow = H + (size_t)arow * F;

    for (int kc = 0; kc < F / 4; ++kc) {
        const int k0 = kc * 4;
        // A fragment: 16x4, lane-half selects K pair {0,1} vs {2,3}
        const float2 t = *(const float2*)(hrow + k0 + 2 * half);
        const v2f a = (v2f){t.x, t.y};
        const int p = 2 * kc + half;                // k-pair index for this half
#pragma unroll
        for (int nt = 0; nt < 8; ++nt) {
            const float2 bb = Wlds[p * F + nt * 16 + l];    // {W[k][n], W[k+1][n]}
            const v2f b = (v2f){bb.x, bb.y};
            acc[nt] = __builtin_amdgcn_wmma_f32_16x16x4_f32(
                false, a, false, b, (short)0, acc[nt], false, false);
        }
    }

#pragma unroll
    for (int nt = 0; nt < 8; ++nt) {
#pragma unroll
        for (int i = 0; i < 8; ++i) {
            const int r = m0 + i + 8 * half;        // C layout: M = i + 8*half
            if (r < nrows) Out[(size_t)r * F + nt * 16 + l] = acc[nt][i];
        }
    }
}

// ---------------------------------------------------------------------------
// gather-scale-scatter: one wave per (edge or self-loop), float4 per lane
// ---------------------------------------------------------------------------
__global__ void scatter_k(const float* __restrict__ hlin, const int* __restrict__ ei,
                          const float* __restrict__ norm, float* __restrict__ agg,
                          int E, int total) {
    const int gid  = blockIdx.x * blockDim.x + threadIdx.x;
    const int edge = gid >> 5;
    const int lane = gid & 31;
    if (edge >= total) return;

    int s, d;
    if (edge < E) { s = ei[edge]; d = ei[E + edge]; }
    else          { s = edge - E; d = s; }

    const float  w  = norm[edge];
    const float4 hv = *(const float4*)(hlin + (size_t)s * F + lane * 4);
    float* dst = agg + (size_t)d * F + lane * 4;
    atomicAdd(dst + 0, hv.x * w);
    atomicAdd(dst + 1, hv.y * w);
    atomicAdd(dst + 2, hv.z * w);
    atomicAdd(dst + 3, hv.w * w);
}

// ---------------------------------------------------------------------------
// bias (+ optional relu) in place over [N,128]
// ---------------------------------------------------------------------------
__global__ void bias_act_k(float* __restrict__ h, const float* __restrict__ b,
                           long total, int relu) {
    long idx = (long)blockIdx.x * blockDim.x + threadIdx.x;
    if (idx >= total) return;
    float v = h[idx] + b[idx & (F - 1)];
    if (relu) v = fmaxf(v, 0.0f);
    h[idx] = v;
}

// ---------------------------------------------------------------------------
// output projection 128 -> 40 (cache-resident, VALU) : out = h @ W_out + b
// ---------------------------------------------------------------------------
__global__ void out_gemm_k(const float* __restrict__ h, const float* __restrict__ Wout,
                           const float* __restrict__ bout, float* __restrict__ out,
                           int nrows) {
    long idx = (long)blockIdx.x * blockDim.x + threadIdx.x;
    if (idx >= (long)nrows * NCLS) return;
    const int n = (int)(idx / NCLS);
    const int c = (int)(idx % NCLS);
    const float* hr = h + (size_t)n * F;
    float acc = bout[c];
#pragma unroll 4
    for (int k = 0; k < F; ++k) acc = fmaf(hr[k], Wout[k * NCLS + c], acc);
    out[idx] = acc;
}

// ---------------------------------------------------------------------------
// row-wise log_softmax over 40 classes, one wave32 per row
// ---------------------------------------------------------------------------
__global__ void logsoftmax_k(float* __restrict__ out, int nrows) {
    const int gid  = blockIdx.x * blockDim.x + threadIdx.x;
    const int row  = gid >> 5;
    const int lane = gid & 31;
    if (row >= nrows) return;
    float* r = out + (size_t)row * NCLS;

    float v0 = r[lane];                                   // lanes 0..31 -> c 0..31
    float v1 = (lane < NCLS - 32) ? r[32 + lane] : -INFINITY;

    float m = fmaxf(v0, v1);
    for (int off = 16; off > 0; off >>= 1) m = fmaxf(m, __shfl_xor(m, off, 32));

    float s = expf(v0 - m) + ((lane < NCLS - 32) ? expf(v1 - m) : 0.0f);
    for (int off = 16; off > 0; off >>= 1) s += __shfl_xor(s, off, 32);

    const float lse = m + logf(s);
    r[lane] = v0 - lse;
    if (lane < NCLS - 32) r[32 + lane] = v1 - lse;
}

// ---------------------------------------------------------------------------
// launch
// ---------------------------------------------------------------------------
extern "C" void kernel_launch(void* const* d_in, const int* in_sizes, int n_in,
                              void* d_out, int out_size, void* d_ws, size_t ws_size,
                              hipStream_t stream) {
    const float* x     = (const float*)d_in[0];
    const int*   ei    = (const int*)  d_in[1];
    const float* W_in  = (const float*)d_in[2];
    const float* b_in  = (const float*)d_in[3];
    const float* W1    = (const float*)d_in[4];
    const float* b1    = (const float*)d_in[5];
    const float* W2    = (const float*)d_in[6];
    const float* b2    = (const float*)d_in[7];
    const float* W_out = (const float*)d_in[8];
    const float* b_out = (const float*)d_in[9];
    float*       out   = (float*)d_out;

    const int Nn = in_sizes[0] / F;     // 50000
    const int E  = in_sizes[1] / 2;     // 800000
    const int total = E + Nn;           // edges + self loops

    // workspace carve-up (all regions 16B aligned given these sizes)
    float* ws   = (float*)d_ws;
    float* deg  = ws;
    float* dinv = deg  + ((Nn + 3) & ~3);
    float* norm = dinv + ((Nn + 3) & ~3);
    float* bufA = norm + ((total + 3) & ~3);
    float* bufB = bufA + (size_t)Nn * F;

    const int T = 256;
    const int gN   = (Nn + T - 1) / T;
    const int gE   = (E + T - 1) / T;
    const int gTot = (total + T - 1) / T;
    const int gNF  = (int)(((long)Nn * F + T - 1) / T);
    const int gGemm = (Nn + 63) / 64;
    const int gScat = (total + 7) / 8;            // 1 wave per edge, 8 waves/block
    const int gOut  = (int)(((long)Nn * NCLS + T - 1) / T);
    const int gLsm  = (Nn + 7) / 8;
    const size_t hbytes = (size_t)Nn * F * sizeof(float);

    // --- normalization ---
    init_deg_k <<<gN,   T, 0, stream>>>(deg, Nn);
    deg_edges_k<<<gE,   T, 0, stream>>>(ei, deg, E);
    dinv_k     <<<gN,   T, 0, stream>>>(deg, dinv, Nn);
    norm_k     <<<gTot, T, 0, stream>>>(ei, dinv, norm, E, total);

    // --- layer 1: h = Agg(x @ W_in) + b_in ---
    gemm128_wmma_k<<<gGemm, 128, 0, stream>>>(x, W_in, bufA, Nn);
    hipMemsetAsync(bufB, 0, hbytes, stream);
    scatter_k <<<gScat, T, 0, stream>>>(bufA, ei, norm, bufB, E, total);
    bias_act_k<<<gNF,   T, 0, stream>>>(bufB, b_in, (long)Nn * F, 0);

    // --- layer 2: h = relu(Agg(h @ W1) + b1) ---
    gemm128_wmma_k<<<gGemm, 128, 0, stream>>>(bufB, W1, bufA, Nn);
    hipMemsetAsync(bufB, 0, hbytes, stream);
    scatter_k <<<gScat, T, 0, stream>>>(bufA, ei, norm, bufB, E, total);
    bias_act_k<<<gNF,   T, 0, stream>>>(bufB, b1, (long)Nn * F, 1);

    // --- layer 3: h = relu(Agg(h @ W2) + b2) ---
    gemm128_wmma_k<<<gGemm, 128, 0, stream>>>(bufB, W2, bufA, Nn);
    hipMemsetAsync(bufB, 0, hbytes, stream);
    scatter_k <<<gScat, T, 0, stream>>>(bufA, ei, norm, bufB, E, total);
    bias_act_k<<<gNF,   T, 0, stream>>>(bufB, b2, (long)Nn * F, 1);

    // --- output projection + log_softmax ---
    out_gemm_k  <<<gOut, T, 0, stream>>>(bufB, W_out, b_out, out, Nn);
    logsoftmax_k<<<gLsm, T, 0, stream>>>(out, Nn);
}